// MessagePassingConvolution_29927332119137
// MI455X (gfx1250) — compile-verified
//
#include <hip/hip_runtime.h>
#include <math.h>

// ---------------------------------------------------------------------------
// MI455X (gfx1250) implementation of the e3nn message-passing convolution.
// GEMMs use V_WMMA_F32_16X16X4_F32 (fp32 matrix pipe: the problem is
// L2-bandwidth / atomic bound, ~6.5 GFLOP total, so f32 WMMA is free and
// preserves reference numerics). The edge-gate GEMM stages Wmlp1 into LDS
// once per block via the Tensor Data Mover (tensor_load_to_lds +
// s_wait_tensorcnt), overlapped with the radial-basis phase.
// ---------------------------------------------------------------------------

typedef __attribute__((ext_vector_type(2))) float v2f;
typedef __attribute__((ext_vector_type(8))) float v8f;
typedef __attribute__((ext_vector_type(4))) unsigned int u32x4;
typedef __attribute__((ext_vector_type(4))) int i32x4;
typedef __attribute__((ext_vector_type(8))) int i32x8;

#define N_NODES 10000
#define N_EDGES 160000
#define EB      32            // edges per block in the edge kernel

__device__ __forceinline__ v8f wmma_f32_k4(v8f acc, v2f a, v2f b) {
    // 8-arg form: (neg_a, A, neg_b, B, c_mod, C, reuse_a, reuse_b)
    return __builtin_amdgcn_wmma_f32_16x16x4_f32(
        false, a, false, b, (short)0, acc, false, false);
}

// ---------------------------------------------------------------------------
// Kernel 1: linear_up.  s_up = node_s @ Wup0 / 8 ; v_up = node_v x Wup1 / 8
// grid = (N/16, 4): blockIdx.y = 0 -> scalars, 1..3 -> vector component m.
// 4 waves / block, each wave owns one 16x16 output tile (64 cols = 4 tiles).
// ---------------------------------------------------------------------------
__global__ __launch_bounds__(128) void node_up_kernel(
    const float* __restrict__ node_s, const float* __restrict__ node_v,
    const float* __restrict__ Wup0,   const float* __restrict__ Wup1,
    float* __restrict__ s_up,         float* __restrict__ v_up)
{
    const int lane    = threadIdx.x & 31;
    const int wave    = threadIdx.x >> 5;
    const int rowtile = blockIdx.x * 16;
    const int mat     = blockIdx.y;
    const int coltile = wave * 16;

    const float* A; const float* W; float* Dst;
    int aRow, aElem, dRow, dElem;
    if (mat == 0) { A = node_s;            W = Wup0; Dst = s_up;            aRow = 64;  aElem = 1; dRow = 64;  dElem = 1; }
    else          { int m = mat - 1;
                    A = node_v + m;        W = Wup1; Dst = v_up + m;        aRow = 192; aElem = 3; dRow = 192; dElem = 3; }

    const int arow  = lane & 15;
    const int koff  = (lane >> 4) << 1;   // lanes 16..31 carry K+2,K+3
    const int ncol  = lane & 15;
    const int rbase = (lane >> 4) * 8;

    if (wave == 0) __builtin_prefetch(&W[lane * 128], 0, 1);

    v8f acc = {};
    #pragma unroll 4
    for (int k0 = 0; k0 < 64; k0 += 4) {
        v2f a, b;
        a.x = A[(rowtile + arow) * aRow + (k0 + koff + 0) * aElem];
        a.y = A[(rowtile + arow) * aRow + (k0 + koff + 1) * aElem];
        b.x = W[(k0 + koff + 0) * 64 + coltile + ncol];
        b.y = W[(k0 + koff + 1) * 64 + coltile + ncol];
        acc = wmma_f32_k4(acc, a, b);
    }
    #pragma unroll
    for (int i = 0; i < 8; ++i)
        Dst[(rowtile + rbase + i) * dRow + (coltile + ncol) * dElem] = acc[i] * 0.125f;
}

// ---------------------------------------------------------------------------
// Kernel 2: per-edge radial basis, gate MLP (WMMA, B from LDS via TDM),
// tensor product, scatter.  256 threads = 8 waves handle EB=32 edges.
//
// Dynamic LDS layout (floats), base offset 0 in the group segment:
//   [     0 .. 16383]  sh_w    : Wmlp1 staged by the Tensor Data Mover (64 KB)
//   [ 16384 .. 24575]  sh_mix  : gates, 32x256                         (32 KB)
//   [ 24576 .. 26623]  sh_h    : MLP hidden post-GELU, 32x64            (8 KB)
//   [ 26624 .. 26879]  sh_es   : radial basis, 32x8                     (1 KB)
//   [ 26880 .. 26975]  sh_y1   : sqrt(3)*vec/r, 32x3
//   [ 26976 .. 27039]  sh_send / sh_recv (int)
// total = 27040 floats = 108160 bytes
// ---------------------------------------------------------------------------
#define EDGE_LDS_BYTES (27040 * 4)

__global__ __launch_bounds__(256) void edge_kernel(
    const float* __restrict__ pos,
    const float* __restrict__ s_up,  const float* __restrict__ v_up,
    const float* __restrict__ Wmlp0, const float* __restrict__ Wmlp1,
    const int*   __restrict__ senders, const int* __restrict__ receivers,
    const int*   __restrict__ cutoff_p,
    float* __restrict__ agg_s, float* __restrict__ agg_v)
{
    extern __shared__ float smem[];
    float* sh_w    = smem;                 // 16384 floats, LDS byte offset 0
    float* sh_mix  = smem + 16384;         //  8192 floats
    float* sh_h    = smem + 24576;         //  2048 floats
    float* sh_es   = smem + 26624;         //   256 floats
    float* sh_y1   = smem + 26880;         //    96 floats
    int*   sh_send = (int*)(smem + 26976); //    32 ints
    int*   sh_recv = (int*)(smem + 27008); //    32 ints

    const int t  = threadIdx.x;
    const int e0 = blockIdx.x * EB;
    const float c = (float)(*cutoff_p);

    // ---- TDM: DMA Wmlp1 (64x256 f32 = 64KB) into LDS, one issue per block --
    // D# per CDNA5 ISA ch.8: 1D tile of 16384 dwords, lds_addr = 0.
    if (t < 32) {   // wave 0 only (TDM is a per-wave scalar op; EXEC ignored)
        const unsigned long long ga = (unsigned long long)(const void*)Wmlp1;
        u32x4 g0 = {};
        g0[0] = 1u;                                        // count=1, user mode
        g0[1] = 0u;                                        // lds_addr = 0
        g0[2] = (unsigned)(ga & 0xFFFFFFFFu);              // global_addr[31:0]
        g0[3] = (unsigned)((ga >> 32) & 0x01FFFFFFu)       // global_addr[56:32]
              | (2u << 30);                                // type = 2 ("image")
        i32x8 g1 = {};
        g1[0] = (2 << 16);                                 // data_size = 4B
        g1[1] = (int)(16384u << 16);                       // tensor_dim0[15:0]
        g1[2] = (int)((16384u >> 16) | (1u << 16));        // dim0[31:16], dim1=1
        g1[3] = (int)(16384u << 16);                       // tile_dim0 = 16384
        g1[4] = 0;                                         // tile_dim1/2 unused
        g1[5] = 16384;                                     // tensor_dim0_stride
        g1[6] = 0; g1[7] = 0;
        i32x4 g2 = {}; i32x4 g3 = {};
#if defined(__clang_major__) && (__clang_major__ >= 23)
        i32x8 g4 = {};
        __builtin_amdgcn_tensor_load_to_lds(g0, g1, g2, g3, g4, 0);
#else
        __builtin_amdgcn_tensor_load_to_lds(g0, g1, g2, g3, 0);
#endif
    }

    // ---- phase 1: edge geometry + bessel * soft_envelope (overlaps DMA) ----
    if (t < EB) {
        const int e = e0 + t;
        const int s = senders[e], r_ = receivers[e];
        sh_send[t] = s; sh_recv[t] = r_;
        const float dx = pos[r_*3+0] - pos[s*3+0];
        const float dy = pos[r_*3+1] - pos[s*3+1];
        const float dz = pos[r_*3+2] - pos[s*3+2];
        const float r    = sqrtf(dx*dx + dy*dy + dz*dz + 1e-12f);
        const float rinv = 1.0f / r;
        sh_y1[t*3+0] = 1.7320508075688772f * dx * rinv;
        sh_y1[t*3+1] = 1.7320508075688772f * dy * rinv;
        sh_y1[t*3+2] = 1.7320508075688772f * dz * rinv;
        // env = 1.2/sus(2) * sus(2*(1-r/c)),  sus(x)=exp(-1/x) for x>0
        const float xs  = 2.0f * (1.0f - r / c);
        const float env = (xs > 0.0f) ? 1.2f * expf(0.5f) * expf(-1.0f / xs) : 0.0f;
        const float pref = sqrtf(2.0f / c) * rinv * env;
        const float w = 3.14159265358979323846f * r / c;
        #pragma unroll
        for (int k = 0; k < 8; ++k)
            sh_es[t*8 + k] = pref * sinf((float)(k + 1) * w);
    }
    __syncthreads();

    // ---- phase 2a: hidden layer 8 -> 64 + tanh-GELU (tiny; overlaps DMA) ---
    #pragma unroll
    for (int i = 0; i < (EB * 64) / 256; ++i) {
        const int idx = t + i * 256;
        const int e = idx >> 6, j = idx & 63;
        float acc = 0.0f;
        #pragma unroll
        for (int k = 0; k < 8; ++k) acc += sh_es[e*8 + k] * Wmlp0[k*64 + j];
        acc *= 0.35355339059327373f;                       // 1/sqrt(8)
        const float x3 = acc * acc * acc;
        sh_h[e*64 + j] = 0.5f * acc *
            (1.0f + tanhf(0.7978845608028654f * (acc + 0.044715f * x3)));
    }
    // DMA must have landed before anyone reads sh_w (wave 0 owns TENSORcnt;
    // the barrier publishes completion to the other waves).
    __builtin_amdgcn_s_wait_tensorcnt(0);
    __syncthreads();

    // ---- phase 2b: mix = h @ Wmlp1 via WMMA f32 16x16x4, B from LDS --------
    // 32x256 output = 16 col-tiles x 2 row-tiles; each wave owns 2 col-tiles
    // and computes both row tiles per K-step so B is loaded once per 2 WMMAs.
    {
        const int lane  = t & 31;
        const int wave  = t >> 5;
        const int arow  = lane & 15;
        const int koff  = (lane >> 4) << 1;
        const int ncol  = lane & 15;
        const int rbase = (lane >> 4) * 8;
        #pragma unroll
        for (int ti = 0; ti < 2; ++ti) {
            const int ct = (wave * 2 + ti) * 16;
            v8f acc0 = {}, acc1 = {};
            #pragma unroll 4
            for (int k0 = 0; k0 < 64; k0 += 4) {
                v2f b, a0, a1;
                b.x  = sh_w[(k0 + koff + 0) * 256 + ct + ncol];
                b.y  = sh_w[(k0 + koff + 1) * 256 + ct + ncol];
                a0.x = sh_h[(arow     ) * 64 + k0 + koff + 0];
                a0.y = sh_h[(arow     ) * 64 + k0 + koff + 1];
                a1.x = sh_h[(arow + 16) * 64 + k0 + koff + 0];
                a1.y = sh_h[(arow + 16) * 64 + k0 + koff + 1];
                acc0 = wmma_f32_k4(acc0, a0, b);
                acc1 = wmma_f32_k4(acc1, a1, b);
            }
            #pragma unroll
            for (int i = 0; i < 8; ++i) {    // 1/sqrt(64) * 1/sqrt(avg_neigh)
                sh_mix[(rbase + i     ) * 256 + ct + ncol] = acc0[i] * (0.125f * 0.25f);
                sh_mix[(rbase + i + 16) * 256 + ct + ncol] = acc1[i] * (0.125f * 0.25f);
            }
        }
    }
    __syncthreads();

    // ---- phase 3: gather sender feats, tensor product, scatter-add ---------
    const int el = t >> 3;              // edge in block
    const int p  = t & 7;               // channel group (8 channels each)
    const int snd = sh_send[el], rcv = sh_recv[el];
    const float y0 = sh_y1[el*3+0], y1 = sh_y1[el*3+1], y2 = sh_y1[el*3+2];
    #pragma unroll
    for (int i = 0; i < 8; ++i) {
        const int ch = p * 8 + i;
        const float ms = s_up[snd*64 + ch];
        const float m0 = v_up[snd*192 + ch*3 + 0];
        const float m1 = v_up[snd*192 + ch*3 + 1];
        const float m2 = v_up[snd*192 + ch*3 + 2];
        const float tps = (m0*y0 + m1*y1 + m2*y2) * 0.5773502691896258f; // 1/sqrt(3)
        const float g0 = sh_mix[el*256 +   0 + ch];
        const float g1 = sh_mix[el*256 +  64 + ch];
        const float g2 = sh_mix[el*256 + 128 + ch];
        const float g3 = sh_mix[el*256 + 192 + ch];
        atomicAdd(&agg_s[rcv*128 +      ch], ms  * g0);
        atomicAdd(&agg_s[rcv*128 + 64 + ch], tps * g1);
        atomicAdd(&agg_v[rcv*384 + ch*3 + 0], m0 * g2);
        atomicAdd(&agg_v[rcv*384 + ch*3 + 1], m1 * g2);
        atomicAdd(&agg_v[rcv*384 + ch*3 + 2], m2 * g2);
        const float tv = ms * g3;
        atomicAdd(&agg_v[rcv*384 + (64 + ch)*3 + 0], tv * y0);
        atomicAdd(&agg_v[rcv*384 + (64 + ch)*3 + 1], tv * y1);
        atomicAdd(&agg_v[rcv*384 + (64 + ch)*3 + 2], tv * y2);
    }
}

// ---------------------------------------------------------------------------
// Kernel 3: linear_down.  out_s = agg_s @ Wd0 / sqrt(128); out_v likewise.
// Output layout matches reference return order: [N,64] then [N,64,3].
// ---------------------------------------------------------------------------
__global__ __launch_bounds__(128) void node_down_kernel(
    const float* __restrict__ agg_s, const float* __restrict__ agg_v,
    const float* __restrict__ Wd0,   const float* __restrict__ Wd1,
    float* __restrict__ out)
{
    const int lane    = threadIdx.x & 31;
    const int wave    = threadIdx.x >> 5;
    const int rowtile = blockIdx.x * 16;
    const int mat     = blockIdx.y;
    const int coltile = wave * 16;

    const float* A; const float* W; float* Dst;
    int aRow, aElem, dRow, dElem;
    if (mat == 0) { A = agg_s;     W = Wd0; Dst = out;                          aRow = 128; aElem = 1; dRow = 64;  dElem = 1; }
    else          { int m = mat - 1;
                    A = agg_v + m; W = Wd1; Dst = out + (size_t)N_NODES*64 + m; aRow = 384; aElem = 3; dRow = 192; dElem = 3; }

    const int arow  = lane & 15;
    const int koff  = (lane >> 4) << 1;
    const int ncol  = lane & 15;
    const int rbase = (lane >> 4) * 8;

    if (wave == 0) __builtin_prefetch(&W[lane * 256], 0, 1);

    v8f acc = {};
    #pragma unroll 4
    for (int k0 = 0; k0 < 128; k0 += 4) {
        v2f a, b;
        a.x = A[(rowtile + arow) * aRow + (k0 + koff + 0) * aElem];
        a.y = A[(rowtile + arow) * aRow + (k0 + koff + 1) * aElem];
        b.x = W[(k0 + koff + 0) * 64 + coltile + ncol];
        b.y = W[(k0 + koff + 1) * 64 + coltile + ncol];
        acc = wmma_f32_k4(acc, a, b);
    }
    #pragma unroll
    for (int i = 0; i < 8; ++i)   // 1/sqrt(128)
        Dst[(rowtile + rbase + i) * dRow + (coltile + ncol) * dElem] = acc[i] * 0.08838834764831845f;
}

// ---------------------------------------------------------------------------
extern "C" void kernel_launch(void* const* d_in, const int* in_sizes, int n_in,
                              void* d_out, int out_size, void* d_ws, size_t ws_size,
                              hipStream_t stream)
{
    const float* pos       = (const float*)d_in[0];
    const float* node_s    = (const float*)d_in[1];
    const float* node_v    = (const float*)d_in[2];
    const float* Wup0      = (const float*)d_in[3];
    const float* Wup1      = (const float*)d_in[4];
    const float* Wmlp0     = (const float*)d_in[5];
    const float* Wmlp1     = (const float*)d_in[6];
    const float* Wd0       = (const float*)d_in[7];
    const float* Wd1       = (const float*)d_in[8];
    const int*   senders   = (const int*)d_in[9];
    const int*   receivers = (const int*)d_in[10];
    const int*   cutoff    = (const int*)d_in[11];

    // Workspace layout (floats): s_up[N*64] | v_up[N*192] | agg_s[N*128] | agg_v[N*384]
    float* s_up  = (float*)d_ws;
    float* v_up  = s_up  + (size_t)N_NODES * 64;
    float* agg_s = v_up  + (size_t)N_NODES * 192;
    float* agg_v = agg_s + (size_t)N_NODES * 128;

    // Zero the aggregation buffers every call (graph-capture legal).
    hipMemsetAsync(agg_s, 0, (size_t)N_NODES * (128 + 384) * sizeof(float), stream);

    node_up_kernel<<<dim3(N_NODES / 16, 4), 128, 0, stream>>>(
        node_s, node_v, Wup0, Wup1, s_up, v_up);

    edge_kernel<<<dim3(N_EDGES / EB), 256, EDGE_LDS_BYTES, stream>>>(
        pos, s_up, v_up, Wmlp0, Wmlp1, senders, receivers, cutoff, agg_s, agg_v);

    node_down_kernel<<<dim3(N_NODES / 16, 4), 128, 0, stream>>>(
        agg_s, agg_v, Wd0, Wd1, (float*)d_out);

    (void)in_sizes; (void)n_in; (void)out_size; (void)ws_size;
}